// SiameseDeltaGNN_46273977647353
// MI455X (gfx1250) — compile-verified
//
#include <hip/hip_runtime.h>
#include <math.h>

// Problem constants (match reference)
#define NODES  20000
#define EDGES  320000
#define ETOT   (EDGES + NODES)   // + self loops
#define FIN    50
#define FINPAD 52                // FIN rounded up to multiple of 4 (WMMA K-step)
#define HID    128
#define NHEADS 4
#define OUTF   (NHEADS * HID)    // 512
#define NGRAPH 32
#define NEG_SLOPE 0.2f

typedef __attribute__((ext_vector_type(2))) float v2f;
typedef __attribute__((ext_vector_type(8))) float v8f;

// ---------------------------------------------------------------------------
// Order-preserving float <-> uint mapping for atomic max on floats
// ---------------------------------------------------------------------------
__device__ __forceinline__ unsigned f2ord(float f) {
    unsigned u = __float_as_uint(f);
    return (u & 0x80000000u) ? ~u : (u | 0x80000000u);
}
__device__ __forceinline__ float ord2f(unsigned u) {
    return (u & 0x80000000u) ? __uint_as_float(u & 0x7fffffffu)
                             : __uint_as_float(~u);
}

// ---------------------------------------------------------------------------
// Padding kernels (layer-1 operands -> K multiple of 4, branch-free GEMM)
// ---------------------------------------------------------------------------
__global__ void pad_x(const float* __restrict__ x, float* __restrict__ xpad) {
    int idx = (int)(blockIdx.x * blockDim.x + threadIdx.x);
    if (idx >= NODES * FINPAD) return;
    int r = idx / FINPAD, c = idx - r * FINPAD;
    xpad[idx] = (c < FIN) ? x[(size_t)r * FIN + c] : 0.0f;
}

__global__ void pad_w1(const float* __restrict__ w, float* __restrict__ wpad) {
    int idx = (int)(blockIdx.x * blockDim.x + threadIdx.x);
    if (idx >= FINPAD * OUTF) return;      // [52,512]; rows >= 50 are zero
    wpad[idx] = (idx < FIN * OUTF) ? w[idx] : 0.0f;
}

// ---------------------------------------------------------------------------
// GEMM: C[M,512] = A[M,K] @ B[K,512], fp32 via V_WMMA_F32_16X16X4_F32.
// K and NCOL are compile-time -> fully unrolled, all loads/stores are
// base + immediate-offset (24-bit IOFFSET), no per-iteration address math.
// Each wave computes TWO stacked 16x16 M-tiles sharing the B fragment.
//
// A frag (16x4 f32): lane l holds M=l&15; half (l>>4) holds K pair {kh,kh+1}.
// B frag (4x16 f32): lane l holds N=l&15; vgpr j holds K = (l>>4)*2 + j.
// C/D (16x16 f32):   lane l, vgpr r -> M = r + (l>>4)*8, N = l&15.
// ---------------------------------------------------------------------------
template <int K>
__global__ void gemm_wmma_f32(const float* __restrict__ A,
                              const float* __restrict__ B,
                              float* __restrict__ C) {
    constexpr int NCOL   = OUTF;            // 512
    constexpr int TILESN = NCOL / 16;       // 32
    constexpr int WAVES  = (NODES / 32) * TILESN;  // 20000

    int wave = (int)((blockIdx.x * blockDim.x + threadIdx.x) >> 5);
    int lane = (int)(threadIdx.x & 31);
    if (wave >= WAVES) return;              // whole-wave uniform exit

    int tm    = wave >> 5;                  // 32-row block index (TILESN==32)
    int tn    = wave & 31;
    int m16   = lane & 15;
    int khalf = (lane >> 4) << 1;           // 0 or 2
    int colB  = (tn << 4) + m16;

    const float* Ap0 = A + (size_t)((tm << 5) + m16) * K + khalf;
    const float* Ap1 = Ap0 + 16 * K;
    const float* Bp  = B + (size_t)khalf * NCOL + colB;
    float*       Cp  = C + (size_t)((tm << 5) + ((lane >> 4) << 3)) * NCOL + colB;

    v8f acc0 = {}, acc1 = {};
#pragma unroll
    for (int k0 = 0; k0 < K; k0 += 4) {
        v2f a0 = *(const v2f*)(Ap0 + k0);   // global_load_b64, imm offset
        v2f a1 = *(const v2f*)(Ap1 + k0);   // global_load_b64, imm offset
        v2f b;
        b.x = Bp[k0 * NCOL];                // global_load_b32, imm offset
        b.y = Bp[k0 * NCOL + NCOL];
        acc0 = __builtin_amdgcn_wmma_f32_16x16x4_f32(
                   false, a0, false, b, (short)0, acc0, false, false);
        acc1 = __builtin_amdgcn_wmma_f32_16x16x4_f32(
                   false, a1, false, b, (short)0, acc1, false, false);
    }

#pragma unroll
    for (int r = 0; r < 8; ++r) {
        Cp[r * NCOL]        = acc0[r];
        Cp[(r + 16) * NCOL] = acc1[r];
    }
}

// ---------------------------------------------------------------------------
// Per-(node,head) attention logits: al = <h[n,hd,:], a[hd,:]>
// ---------------------------------------------------------------------------
__global__ void attn_logits(const float* __restrict__ h,
                            const float* __restrict__ a_src,
                            const float* __restrict__ a_dst,
                            float* __restrict__ al_s,
                            float* __restrict__ al_d) {
    int idx = (int)(blockIdx.x * blockDim.x + threadIdx.x);
    if (idx >= NODES * NHEADS) return;
    int n  = idx >> 2;
    int hd = idx & 3;
    const float* hp = h + ((size_t)n * NHEADS + hd) * HID;
    const float* as = a_src + (size_t)hd * HID;
    const float* ad = a_dst + (size_t)hd * HID;
    float s = 0.0f, d = 0.0f;
#pragma unroll 8
    for (int i = 0; i < HID; ++i) {
        float v = hp[i];
        s += v * as[i];
        d += v * ad[i];
    }
    al_s[idx] = s;
    al_d[idx] = d;
}

__global__ void fill_f32(float* __restrict__ p, float v, int n) {
    int idx = (int)(blockIdx.x * blockDim.x + threadIdx.x);
    if (idx < n) p[idx] = v;
}

// ---------------------------------------------------------------------------
// Edge pass 1: e = leaky_relu(al_src[src] + al_dst[dst]); segment max via
// order-preserving uint atomic max.
// ---------------------------------------------------------------------------
__global__ void edge_logits_max(const int* __restrict__ src,
                                const int* __restrict__ dst,
                                const float* __restrict__ al_s,
                                const float* __restrict__ al_d,
                                float* __restrict__ ev,
                                unsigned* __restrict__ emax) {
    int idx = (int)(blockIdx.x * blockDim.x + threadIdx.x);
    if (idx >= ETOT * NHEADS) return;
    int e  = idx >> 2;
    int hd = idx & 3;
    int s = (e < EDGES) ? src[e] : (e - EDGES);
    int d = (e < EDGES) ? dst[e] : (e - EDGES);
    float v = al_s[s * NHEADS + hd] + al_d[d * NHEADS + hd];
    v = (v > 0.0f) ? v : NEG_SLOPE * v;
    ev[idx] = v;
    atomicMax(&emax[d * NHEADS + hd], f2ord(v));
}

// Edge pass 2: e_exp = exp(e - max[dst]); segment sum.
__global__ void edge_expsum(const int* __restrict__ src,
                            const int* __restrict__ dst,
                            float* __restrict__ ev,
                            const unsigned* __restrict__ emax,
                            float* __restrict__ esum) {
    int idx = (int)(blockIdx.x * blockDim.x + threadIdx.x);
    if (idx >= ETOT * NHEADS) return;
    int e  = idx >> 2;
    int hd = idx & 3;
    int d = (e < EDGES) ? dst[e] : (e - EDGES);
    float m = ord2f(emax[d * NHEADS + hd]);
    float x = expf(ev[idx] - m);
    ev[idx] = x;
    atomicAdd(&esum[d * NHEADS + hd], x);
}

// Edge pass 3: acc[dst,hd,:] += h[src,hd,:] * alpha
__global__ void edge_aggregate(const int* __restrict__ src,
                               const int* __restrict__ dst,
                               const float* __restrict__ ev,
                               const float* __restrict__ esum,
                               const float* __restrict__ h,
                               float* __restrict__ acc) {
    int idx = (int)(blockIdx.x * blockDim.x + threadIdx.x);
    if (idx >= ETOT * NHEADS) return;
    int e  = idx >> 2;
    int hd = idx & 3;
    int s = (e < EDGES) ? src[e] : (e - EDGES);
    int d = (e < EDGES) ? dst[e] : (e - EDGES);
    float alpha = ev[idx] / (esum[d * NHEADS + hd] + 1e-16f);
    const float* hp = h   + ((size_t)s * NHEADS + hd) * HID;
    float*       op = acc + ((size_t)d * NHEADS + hd) * HID;
    __builtin_prefetch(hp, 0, 0);   // global_prefetch_b8: pull src row to WGP
#pragma unroll 4
    for (int i = 0; i < HID; i += 4) {
        float4 v = *(const float4*)(hp + i);
        atomicAdd(op + i + 0, v.x * alpha);
        atomicAdd(op + i + 1, v.y * alpha);
        atomicAdd(op + i + 2, v.z * alpha);
        atomicAdd(op + i + 3, v.w * alpha);
    }
}

// Head mean + bias + ReLU (both GAT layers are followed by ReLU in encode)
__global__ void head_mean_relu(const float* __restrict__ acc,
                               const float* __restrict__ bias,
                               float* __restrict__ x_out) {
    int idx = (int)(blockIdx.x * blockDim.x + threadIdx.x);
    if (idx >= NODES * HID) return;
    int n  = idx >> 7;          // / HID
    int dm = idx & 127;         // % HID
    float s = 0.0f;
#pragma unroll
    for (int hd = 0; hd < NHEADS; ++hd)
        s += acc[((size_t)n * NHEADS + hd) * HID + dm];
    float v = s * 0.25f + bias[dm];
    x_out[(size_t)n * HID + dm] = fmaxf(v, 0.0f);
}

// global_mean_pool: sum + counts
__global__ void pool_sum(const float* __restrict__ emb,
                         const int* __restrict__ batch,
                         float* __restrict__ gsum,
                         float* __restrict__ gcnt) {
    int idx = (int)(blockIdx.x * blockDim.x + threadIdx.x);
    if (idx >= NODES * HID) return;
    int n  = idx >> 7;
    int dm = idx & 127;
    int g  = batch[n];
    atomicAdd(&gsum[(size_t)g * HID + dm], emb[idx]);
    if (dm == 0) atomicAdd(&gcnt[g], 1.0f);
}

__global__ void pool_div(float* __restrict__ gsum,
                         const float* __restrict__ gcnt) {
    int idx = (int)(blockIdx.x * blockDim.x + threadIdx.x);
    if (idx >= NGRAPH * HID) return;
    int g = idx >> 7;
    gsum[idx] /= fmaxf(gcnt[g], 1.0f);
}

// Head MLP: out[g] = relu(concat(mt-wt, esm) @ fc1 + b1) @ fc2 + b2
__global__ void mlp_head(const float* __restrict__ wt_emb,
                         const float* __restrict__ mt_emb,
                         const float* __restrict__ esm,
                         const float* __restrict__ fc1w,  // [HID+FIN, 64]
                         const float* __restrict__ fc1b,
                         const float* __restrict__ fc2w,  // [64, 1]
                         const float* __restrict__ fc2b,
                         float* __restrict__ outp) {
    int g = (int)blockIdx.x;
    int j = (int)threadIdx.x;   // 64 threads
    float s = fc1b[j];
    for (int k = 0; k < HID; ++k)
        s += (mt_emb[(size_t)g * HID + k] - wt_emb[(size_t)g * HID + k]) * fc1w[k * 64 + j];
    for (int k = 0; k < FIN; ++k)
        s += esm[(size_t)g * FIN + k] * fc1w[(HID + k) * 64 + j];
    __shared__ float red[64];
    red[j] = fmaxf(s, 0.0f) * fc2w[j];
    __syncthreads();
    for (int off = 32; off > 0; off >>= 1) {
        if (j < off) red[j] += red[j + off];
        __syncthreads();
    }
    if (j == 0) outp[g] = red[0] + fc2b[0];
}

// ---------------------------------------------------------------------------
// Host-side orchestration
// ---------------------------------------------------------------------------
static inline int nblk(long n, int b) { return (int)((n + b - 1) / b); }

template <int K>
static void run_gat_layer(const float* x_in,
                          const float* W, const float* a_s, const float* a_d,
                          const float* bias,
                          const int* src, const int* dst,
                          float* h, float* acc,
                          float* al_s, float* al_d,
                          unsigned* emax, float* esum, float* ev,
                          float* x_out, hipStream_t stream) {
    const long waves = (long)(NODES / 32) * (OUTF / 16);  // 20000
    gemm_wmma_f32<K><<<nblk(waves * 32, 256), 256, 0, stream>>>(x_in, W, h);
    attn_logits<<<nblk((long)NODES * NHEADS, 256), 256, 0, stream>>>(h, a_s, a_d, al_s, al_d);
    fill_f32<<<nblk((long)NODES * OUTF, 256), 256, 0, stream>>>(acc, 0.0f, NODES * OUTF);
    fill_f32<<<nblk((long)NODES * NHEADS, 256), 256, 0, stream>>>((float*)emax, 0.0f, NODES * NHEADS); // 0u == identity of f2ord-max
    fill_f32<<<nblk((long)NODES * NHEADS, 256), 256, 0, stream>>>(esum, 0.0f, NODES * NHEADS);
    edge_logits_max<<<nblk((long)ETOT * NHEADS, 256), 256, 0, stream>>>(src, dst, al_s, al_d, ev, emax);
    edge_expsum<<<nblk((long)ETOT * NHEADS, 256), 256, 0, stream>>>(src, dst, ev, emax, esum);
    edge_aggregate<<<nblk((long)ETOT * NHEADS, 256), 256, 0, stream>>>(src, dst, ev, esum, h, acc);
    head_mean_relu<<<nblk((long)NODES * HID, 256), 256, 0, stream>>>(acc, bias, x_out);
}

static void encode_graph(const float* x, const int* ei, const int* batch,
                         const float* W1pad, const float* as1, const float* ad1, const float* b1,
                         const float* W2, const float* as2, const float* ad2, const float* b2,
                         float* xpad, float* h, float* acc, float* x2,
                         float* al_s, float* al_d, unsigned* emax, float* esum, float* ev,
                         float* emb, float* gcnt, hipStream_t stream) {
    const int* src = ei;            // edge_index[0,:]
    const int* dst = ei + EDGES;    // edge_index[1,:]
    pad_x<<<nblk((long)NODES * FINPAD, 256), 256, 0, stream>>>(x, xpad);
    run_gat_layer<FINPAD>(xpad, W1pad, as1, ad1, b1, src, dst,
                          h, acc, al_s, al_d, emax, esum, ev, x2, stream);
    run_gat_layer<HID>(x2, W2, as2, ad2, b2, src, dst,
                       h, acc, al_s, al_d, emax, esum, ev, x2, stream);
    fill_f32<<<nblk((long)NGRAPH * HID, 256), 256, 0, stream>>>(emb, 0.0f, NGRAPH * HID);
    fill_f32<<<1, 256, 0, stream>>>(gcnt, 0.0f, NGRAPH);
    pool_sum<<<nblk((long)NODES * HID, 256), 256, 0, stream>>>(x2, batch, emb, gcnt);
    pool_div<<<nblk((long)NGRAPH * HID, 256), 256, 0, stream>>>(emb, gcnt);
}

extern "C" void kernel_launch(void* const* d_in, const int* in_sizes, int n_in,
                              void* d_out, int out_size, void* d_ws, size_t ws_size,
                              hipStream_t stream) {
    const float* wt_x     = (const float*)d_in[0];
    const int*   wt_ei    = (const int*)  d_in[1];
    const int*   wt_batch = (const int*)  d_in[3];
    const float* mt_x     = (const float*)d_in[4];
    const int*   mt_ei    = (const int*)  d_in[5];
    const int*   mt_batch = (const int*)  d_in[7];
    const float* esm      = (const float*)d_in[8];
    const float* W1  = (const float*)d_in[9];
    const float* as1 = (const float*)d_in[10];
    const float* ad1 = (const float*)d_in[11];
    const float* b1  = (const float*)d_in[12];
    const float* W2  = (const float*)d_in[13];
    const float* as2 = (const float*)d_in[14];
    const float* ad2 = (const float*)d_in[15];
    const float* b2  = (const float*)d_in[16];
    const float* fc1w = (const float*)d_in[17];
    const float* fc1b = (const float*)d_in[18];
    const float* fc2w = (const float*)d_in[19];
    const float* fc2b = (const float*)d_in[20];

    // Workspace carve-up (~104 MB total, fits the 192 MB L2)
    float* ws = (float*)d_ws;
    float*    h      = ws; ws += (size_t)NODES * OUTF;     // 40.96 MB
    float*    acc    = ws; ws += (size_t)NODES * OUTF;     // 40.96 MB
    float*    x2     = ws; ws += (size_t)NODES * HID;      // 10.24 MB
    float*    xpad   = ws; ws += (size_t)NODES * FINPAD;   //  4.16 MB
    float*    W1pad  = ws; ws += (size_t)FINPAD * OUTF;    //  0.10 MB
    float*    al_s   = ws; ws += (size_t)NODES * NHEADS;
    float*    al_d   = ws; ws += (size_t)NODES * NHEADS;
    unsigned* emax   = (unsigned*)ws; ws += (size_t)NODES * NHEADS;
    float*    esum   = ws; ws += (size_t)NODES * NHEADS;
    float*    ev     = ws; ws += (size_t)ETOT * NHEADS;    //  5.44 MB
    float*    wt_emb = ws; ws += (size_t)NGRAPH * HID;
    float*    mt_emb = ws; ws += (size_t)NGRAPH * HID;
    float*    gcnt   = ws; ws += NGRAPH;

    // Pad W1 once (shared by both encodes)
    pad_w1<<<nblk((long)FINPAD * OUTF, 256), 256, 0, stream>>>(W1, W1pad);

    encode_graph(wt_x, wt_ei, wt_batch, W1pad, as1, ad1, b1, W2, as2, ad2, b2,
                 xpad, h, acc, x2, al_s, al_d, emax, esum, ev, wt_emb, gcnt, stream);
    encode_graph(mt_x, mt_ei, mt_batch, W1pad, as1, ad1, b1, W2, as2, ad2, b2,
                 xpad, h, acc, x2, al_s, al_d, emax, esum, ev, mt_emb, gcnt, stream);

    mlp_head<<<NGRAPH, 64, 0, stream>>>(wt_emb, mt_emb, esm,
                                        fc1w, fc1b, fc2w, fc2b, (float*)d_out);
}